// Seq2SeqAttentionDecoder_60739427501035
// MI455X (gfx1250) — compile-verified
//
#include <hip/hip_runtime.h>
#include <hip/hip_bf16.h>

// Problem constants (from reference)
#define B_   64
#define T_   64
#define S_   64
#define H_   512
#define E_   256
#define V_   32000
#define G3H  (3 * H_)    // 1536
#define KIH0 (H_ + E_)   // 768

#define KC   32          // K-chunk staged in LDS
#define LDP  36          // padded LDS row stride (floats): 16B-aligned, no bank conflicts

typedef float v2f __attribute__((ext_vector_type(2)));
typedef float v8f __attribute__((ext_vector_type(8)));
typedef int   v4i __attribute__((ext_vector_type(4)));

static __device__ inline v8f vzero8() {
  v8f z = {0.f, 0.f, 0.f, 0.f, 0.f, 0.f, 0.f, 0.f};
  return z;
}

// D = A(16x4) * B(4x16) + C, exact-fp32 WMMA.
static __device__ inline v8f wmma4(v2f a, v2f b, v8f c) {
  return __builtin_amdgcn_wmma_f32_16x16x4_f32(
      false, a, false, b, (short)0, c, false, false);
}

// Async global->LDS copy support (codegen-probed: 4 args, param0 = AS1 int4*)
#if defined(__has_builtin)
#if __has_builtin(__builtin_amdgcn_global_load_async_to_lds_b128) && \
    __has_builtin(__builtin_amdgcn_s_wait_asynccnt)
#define HAVE_ASYNC_LDS 1
#endif
#endif

#ifdef HAVE_ASYNC_LDS
typedef __attribute__((address_space(1))) v4i* gp128;
typedef __attribute__((address_space(3))) v4i* lp128;
static __device__ inline void async_cp16(const float* gsrc, float* ldst) {
  __builtin_amdgcn_global_load_async_to_lds_b128((gp128)gsrc, (lp128)ldst, 0, 0);
}
#endif

// ---------------------------------------------------------------------------
// Direct (register) 16x64 strip, software-pipelined: preload next k-step's
// fragments before issuing the current WMMAs. Used by the small,
// latency-bound per-step GRU GEMMs.
// ---------------------------------------------------------------------------
static __device__ inline void tile16x64(const float* __restrict__ A, int lda,
                                        const float* __restrict__ W, int ldw,
                                        int row0, int col0, int K, v8f acc[4]) {
  const int lane = threadIdx.x & 31;
  const int n    = lane & 15;
  const int koff = (lane >> 4) * 2;
  const float* ar = A + (size_t)(row0 + n) * lda + koff;
  const float* w0 = W + (size_t)(col0 +  0 + n) * ldw + koff;
  const float* w1 = W + (size_t)(col0 + 16 + n) * ldw + koff;
  const float* w2 = W + (size_t)(col0 + 32 + n) * ldw + koff;
  const float* w3 = W + (size_t)(col0 + 48 + n) * ldw + koff;

  v2f a  = *(const v2f*)(ar);
  v2f b0 = *(const v2f*)(w0);
  v2f b1 = *(const v2f*)(w1);
  v2f b2 = *(const v2f*)(w2);
  v2f b3 = *(const v2f*)(w3);
#pragma unroll 4
  for (int k = 0; k + 4 < K; k += 4) {
    v2f an  = *(const v2f*)(ar + k + 4);
    v2f bn0 = *(const v2f*)(w0 + k + 4);
    v2f bn1 = *(const v2f*)(w1 + k + 4);
    v2f bn2 = *(const v2f*)(w2 + k + 4);
    v2f bn3 = *(const v2f*)(w3 + k + 4);
    acc[0] = wmma4(a, b0, acc[0]);
    acc[1] = wmma4(a, b1, acc[1]);
    acc[2] = wmma4(a, b2, acc[2]);
    acc[3] = wmma4(a, b3, acc[3]);
    a = an; b0 = bn0; b1 = bn1; b2 = bn2; b3 = bn3;
  }
  acc[0] = wmma4(a, b0, acc[0]);
  acc[1] = wmma4(a, b1, acc[1]);
  acc[2] = wmma4(a, b2, acc[2]);
  acc[3] = wmma4(a, b3, acc[3]);
}

// ---------------------------------------------------------------------------
// LDS-staged, double-buffered GEMM for K=512: C = A[M,512] @ W[N,512]^T.
// Block: 256 threads (8 waves), output tile 128 rows x 64 cols.
// grid(N/64, M/128). remap=1 writes [B,T,V] from row index m = t*B + b.
// Chunk c+1 is copied global->LDS with ASYNC b128 copies (ASYNCcnt) while
// chunk c computes; s_wait_asynccnt 0 + barrier per chunk.
// ---------------------------------------------------------------------------
__global__ __launch_bounds__(256) void gemm512_lds_kernel(
    const float* __restrict__ A, const float* __restrict__ W,
    const float* __restrict__ bias, float* __restrict__ C,
    int N, int remap) {
  __shared__ float As[2][128 * LDP];   // 36 KB
  __shared__ float Ws[2][64 * LDP];    // 18 KB
  const int tid  = threadIdx.x;
  const int wave = tid >> 5;
  const int lane = tid & 31;
  const int n = lane & 15, half = lane >> 4;
  const int rowBase = blockIdx.y * 128;
  const int colBase = blockIdx.x * 64;

#ifdef HAVE_ASYNC_LDS
  auto issueChunk = [&](int kc, int buf) {
#pragma unroll
    for (int i = 0; i < 4; ++i) {                 // A: 128 rows x 32 floats
      int idx = tid + i * 256;                    // 0..1023 float4s
      int r = idx >> 3, c = (idx & 7) * 4;
      async_cp16(A + (size_t)(rowBase + r) * H_ + kc + c,
                 &As[buf][r * LDP + c]);
    }
#pragma unroll
    for (int i = 0; i < 2; ++i) {                 // W: 64 rows x 32 floats
      int idx = tid + i * 256;                    // 0..511 float4s
      int r = idx >> 3, c = (idx & 7) * 4;
      async_cp16(W + (size_t)(colBase + r) * H_ + kc + c,
                 &Ws[buf][r * LDP + c]);
    }
  };
#else
  float4 aReg[4];
  float4 wReg[2];
  auto loadChunk = [&](int kc) {
#pragma unroll
    for (int i = 0; i < 4; ++i) {
      int idx = tid + i * 256;
      int r = idx >> 3, c = (idx & 7) * 4;
      aReg[i] = *(const float4*)(A + (size_t)(rowBase + r) * H_ + kc + c);
    }
#pragma unroll
    for (int i = 0; i < 2; ++i) {
      int idx = tid + i * 256;
      int r = idx >> 3, c = (idx & 7) * 4;
      wReg[i] = *(const float4*)(W + (size_t)(colBase + r) * H_ + kc + c);
    }
  };
  auto storeChunk = [&](int buf) {
#pragma unroll
    for (int i = 0; i < 4; ++i) {
      int idx = tid + i * 256;
      int r = idx >> 3, c = (idx & 7) * 4;
      *(float4*)&As[buf][r * LDP + c] = aReg[i];
    }
#pragma unroll
    for (int i = 0; i < 2; ++i) {
      int idx = tid + i * 256;
      int r = idx >> 3, c = (idx & 7) * 4;
      *(float4*)&Ws[buf][r * LDP + c] = wReg[i];
    }
  };
#endif

  v8f acc[4] = {vzero8(), vzero8(), vzero8(), vzero8()};

#ifdef HAVE_ASYNC_LDS
  issueChunk(0, 0);
  __builtin_amdgcn_s_wait_asynccnt(0);
  __syncthreads();
#else
  loadChunk(0);
  storeChunk(0);
  __syncthreads();
#endif

  const int NCHUNK = H_ / KC;                     // 16
#pragma unroll 1
  for (int cchunk = 0; cchunk < NCHUNK; ++cchunk) {
#ifdef HAVE_ASYNC_LDS
    if (cchunk + 1 < NCHUNK) issueChunk((cchunk + 1) * KC, (cchunk + 1) & 1);
#else
    if (cchunk + 1 < NCHUNK) loadChunk((cchunk + 1) * KC);
#endif

    const float* ap  = &As[cchunk & 1][(wave * 16 + n) * LDP + 2 * half];
    const float* wp0 = &Ws[cchunk & 1][( 0 + n) * LDP + 2 * half];
    const float* wp1 = &Ws[cchunk & 1][(16 + n) * LDP + 2 * half];
    const float* wp2 = &Ws[cchunk & 1][(32 + n) * LDP + 2 * half];
    const float* wp3 = &Ws[cchunk & 1][(48 + n) * LDP + 2 * half];
#pragma unroll
    for (int k = 0; k < KC; k += 4) {
      v2f a  = *(const v2f*)(ap  + k);
      v2f b0 = *(const v2f*)(wp0 + k);
      v2f b1 = *(const v2f*)(wp1 + k);
      v2f b2 = *(const v2f*)(wp2 + k);
      v2f b3 = *(const v2f*)(wp3 + k);
      acc[0] = wmma4(a, b0, acc[0]);
      acc[1] = wmma4(a, b1, acc[1]);
      acc[2] = wmma4(a, b2, acc[2]);
      acc[3] = wmma4(a, b3, acc[3]);
    }
#ifdef HAVE_ASYNC_LDS
    if (cchunk + 1 < NCHUNK) __builtin_amdgcn_s_wait_asynccnt(0);
#else
    if (cchunk + 1 < NCHUNK) storeChunk((cchunk + 1) & 1);
#endif
    __syncthreads();
  }

  const int row0 = rowBase + wave * 16;
#pragma unroll
  for (int tt = 0; tt < 4; ++tt) {
    const int col = colBase + tt * 16 + n;
    const float bv = bias ? bias[col] : 0.f;
#pragma unroll
    for (int r = 0; r < 8; ++r) {
      const int m = row0 + r + 8 * half;
      const float v = acc[tt][r] + bv;
      if (remap) {                                 // m = t*B + b -> [b][t][col]
        const int t = m >> 6, b = m & (B_ - 1);
        C[((size_t)b * T_ + t) * (size_t)N + col] = v;
      } else {
        C[(size_t)m * (size_t)N + col] = v;
      }
    }
  }
}

// ---------------- embedding gather: x_seq[t][b][:] = emb[X[b][t]] ----------
__global__ void embed_kernel(const int* __restrict__ X,
                             const float* __restrict__ emb,
                             float* __restrict__ x_seq) {
  int bt = blockIdx.x;          // b*T + t  (X is [B,T])
  int b = bt / T_, t = bt % T_;
  int tok = X[bt];
  int e = threadIdx.x;          // E_ == blockDim.x == 256
  x_seq[((size_t)t * B_ + b) * E_ + e] = emb[(size_t)tok * E_ + e];
}

// ---------------- attention step (one block per batch row) -----------------
__global__ void attn_kernel(const float* __restrict__ h1,        // [B,H]
                            const float* __restrict__ keys_proj, // [B,S,H]
                            const float* __restrict__ enc,       // [B,S,H]
                            const float* __restrict__ Wq,        // [H,H]
                            const float* __restrict__ wv,        // [H]
                            const int*   __restrict__ vlen,      // [B]
                            const float* __restrict__ x_seq,     // [T,B,E]
                            int t,
                            float* __restrict__ inp) {           // [B,H+E]
  __shared__ float sh_h[H_];
  __shared__ float sh_q[H_];
  __shared__ float sh_sc[S_];
  __shared__ float sh_red[256];
  __shared__ float sh_m, sh_sum;
  const int b = blockIdx.x, tid = threadIdx.x;

  for (int i = tid; i < H_; i += 256) sh_h[i] = h1[b * H_ + i];
  __syncthreads();

  for (int k = tid; k < H_; k += 256) {
    const float* wr = Wq + (size_t)k * H_;
    float s = 0.f;
#pragma unroll 4
    for (int h = 0; h < H_; ++h) s += sh_h[h] * wr[h];
    sh_q[k] = s;
  }
  __syncthreads();

  {
    const int s = tid >> 2, j = tid & 3;
    const float* kp = keys_proj + (size_t)(b * S_ + s) * H_;
    float p = 0.f;
    for (int k = j * 128; k < j * 128 + 128; ++k)
      p += tanhf(sh_q[k] + kp[k]) * wv[k];
    sh_red[tid] = p;
  }
  __syncthreads();
  if (tid < S_) {
    float sc = sh_red[tid * 4] + sh_red[tid * 4 + 1] +
               sh_red[tid * 4 + 2] + sh_red[tid * 4 + 3];
    sh_sc[tid] = (tid < vlen[b]) ? sc : -1e6f;
  }
  __syncthreads();
  if (tid == 0) {
    float m = sh_sc[0];
    for (int s = 1; s < S_; ++s) m = fmaxf(m, sh_sc[s]);
    sh_m = m;
  }
  __syncthreads();
  if (tid < S_) sh_sc[tid] = __expf(sh_sc[tid] - sh_m);
  __syncthreads();
  if (tid == 0) {
    float s = 0.f;
    for (int i = 0; i < S_; ++i) s += sh_sc[i];
    sh_sum = s;
  }
  __syncthreads();
  const float inv = 1.0f / sh_sum;

  for (int h = tid; h < H_; h += 256) {
    float acc = 0.f;
#pragma unroll 4
    for (int s = 0; s < S_; ++s)
      acc += sh_sc[s] * enc[(size_t)(b * S_ + s) * H_ + h];
    inp[b * KIH0 + h] = acc * inv;
  }
  for (int e = tid; e < E_; e += 256)
    inp[b * KIH0 + H_ + e] = x_seq[((size_t)t * B_ + b) * E_ + e];
}

// ---------------- GRU layer GEMMs: one wave per block, grid(3H/64, B/16) ----
__global__ __launch_bounds__(32) void gru_gemm_kernel(
    const float* __restrict__ X, int Kx,
    const float* __restrict__ Hprev,
    const float* __restrict__ W_ih, const float* __restrict__ b_ih,
    const float* __restrict__ W_hh, const float* __restrict__ b_hh,
    float* __restrict__ gi, float* __restrict__ gh) {
  const int row0 = blockIdx.y * 16;
  const int col0 = blockIdx.x * 64;
  v8f ai[4] = {vzero8(), vzero8(), vzero8(), vzero8()};
  v8f ah[4] = {vzero8(), vzero8(), vzero8(), vzero8()};
  tile16x64(X, Kx, W_ih, Kx, row0, col0, Kx, ai);
  tile16x64(Hprev, H_, W_hh, H_, row0, col0, H_, ah);
  const int lane = threadIdx.x & 31, n = lane & 15, half = lane >> 4;
#pragma unroll
  for (int tt = 0; tt < 4; ++tt)
#pragma unroll
    for (int r = 0; r < 8; ++r) {
      const int row = row0 + r + 8 * half;
      const int col = col0 + tt * 16 + n;
      gi[(size_t)row * G3H + col] = ai[tt][r] + b_ih[col];
      gh[(size_t)row * G3H + col] = ah[tt][r] + b_hh[col];
    }
}

// ---------------- GRU gates (elementwise), in-place h update ---------------
__global__ void gru_gate_kernel(const float* __restrict__ gi,
                                const float* __restrict__ gh,
                                float* __restrict__ h,       // [B,H] in-out
                                float* __restrict__ out_t) { // [B,H] or null
  const int idx = blockIdx.x * blockDim.x + threadIdx.x;     // 0..B*H
  const int m = idx >> 9, j = idx & (H_ - 1);
  const float ir  = gi[(size_t)m * G3H + j];
  const float iz  = gi[(size_t)m * G3H + H_ + j];
  const float in_ = gi[(size_t)m * G3H + 2 * H_ + j];
  const float hr  = gh[(size_t)m * G3H + j];
  const float hz  = gh[(size_t)m * G3H + H_ + j];
  const float hn  = gh[(size_t)m * G3H + 2 * H_ + j];
  const float r = 1.f / (1.f + __expf(-(ir + hr)));
  const float z = 1.f / (1.f + __expf(-(iz + hz)));
  const float nn = tanhf(in_ + r * hn);
  const float hp = h[idx];
  const float hnew = (1.f - z) * nn + z * hp;
  h[idx] = hnew;
  if (out_t) out_t[idx] = hnew;
}

extern "C" void kernel_launch(void* const* d_in, const int* in_sizes, int n_in,
                              void* d_out, int out_size, void* d_ws, size_t ws_size,
                              hipStream_t stream) {
  const int*   X      = (const int*)  d_in[0];
  const float* enc    = (const float*)d_in[1];
  const float* hstate = (const float*)d_in[2];   // [2,B,H]
  const int*   vlen   = (const int*)  d_in[3];
  const float* emb    = (const float*)d_in[4];
  const float* Wq     = (const float*)d_in[5];
  const float* Wk     = (const float*)d_in[6];
  const float* wv     = (const float*)d_in[7];
  const float* W_ih0  = (const float*)d_in[8];
  const float* W_hh0  = (const float*)d_in[9];
  const float* b_ih0  = (const float*)d_in[10];
  const float* b_hh0  = (const float*)d_in[11];
  const float* W_ih1  = (const float*)d_in[12];
  const float* W_hh1  = (const float*)d_in[13];
  const float* b_ih1  = (const float*)d_in[14];
  const float* b_hh1  = (const float*)d_in[15];
  const float* Wd     = (const float*)d_in[16];
  const float* bd     = (const float*)d_in[17];
  float* out = (float*)d_out;

  // workspace layout (~22.2 MB of floats)
  float* ws    = (float*)d_ws;
  float* x_seq = ws;                                  // T*B*E
  float* keys  = x_seq + (size_t)T_ * B_ * E_;        // B*S*H
  float* h0    = keys + (size_t)B_ * S_ * H_;         // B*H
  float* h1    = h0 + (size_t)B_ * H_;                // B*H
  float* inp   = h1 + (size_t)B_ * H_;                // B*(H+E)
  float* gi    = inp + (size_t)B_ * KIH0;             // B*3H
  float* gh    = gi + (size_t)B_ * G3H;               // B*3H
  float* outs  = gh + (size_t)B_ * G3H;               // T*B*H

  // 1) embeddings + initial hidden state
  embed_kernel<<<B_ * T_, E_, 0, stream>>>(X, emb, x_seq);
  (void)hipMemcpyAsync(h0, hstate, (size_t)2 * B_ * H_ * sizeof(float),
                       hipMemcpyDeviceToDevice, stream);

  // 2) keys_proj = enc_outputs @ Wk^T  [4096 x 512], LDS-staged WMMA
  gemm512_lds_kernel<<<dim3(H_ / 64, (B_ * S_) / 128), 256, 0, stream>>>(
      enc, Wk, nullptr, keys, H_, 0);

  // 3) sequential scan over T (stream order gives per-step sync)
  for (int t = 0; t < T_; ++t) {
    attn_kernel<<<B_, 256, 0, stream>>>(h1, keys, enc, Wq, wv, vlen, x_seq, t, inp);
    gru_gemm_kernel<<<dim3(G3H / 64, B_ / 16), 32, 0, stream>>>(
        inp, KIH0, h0, W_ih0, b_ih0, W_hh0, b_hh0, gi, gh);
    gru_gate_kernel<<<(B_ * H_) / 256, 256, 0, stream>>>(gi, gh, h0, nullptr);
    gru_gemm_kernel<<<dim3(G3H / 64, B_ / 16), 32, 0, stream>>>(
        h0, H_, h1, W_ih1, b_ih1, W_hh1, b_hh1, gi, gh);
    gru_gate_kernel<<<(B_ * H_) / 256, 256, 0, stream>>>(gi, gh, h1,
                                                         outs + (size_t)t * B_ * H_);
  }

  // 4) logits = outs @ dense_W^T + b -> [B,T,V]  (store-BW bound, L2-resident W)
  gemm512_lds_kernel<<<dim3(V_ / 64, (T_ * B_) / 128), 256, 0, stream>>>(
      outs, Wd, bd, out, V_, 1);
}